// JointLoss_66554813219369
// MI455X (gfx1250) — compile-verified
//
#include <hip/hip_runtime.h>
#include <hip/hip_bf16.h>
#include <math.h>

typedef __attribute__((ext_vector_type(2))) float v2f;
typedef __attribute__((ext_vector_type(8))) float v8f;

#define BB 32
#define GG 36
#define AN 5
#define TT 8
#define NN (BB * GG * GG * AN)   // 207360
#define LM 136
#define IMG 288.0f
#define L_NOOBJ 0.5f
#define L_COOR 5.0f

struct Ws {
    int   lin[BB * TT];       // cell index per target, NN if invalid
    int   winner[BB * TT];    // 1 if this target owns its cell
    int   objCount;
    int   pad[3];
    float logbox[BB * TT * 4];
    float sumAll;             // sum conf^2 over all N cells
    float sumObjSq;           // sum conf^2 over obj cells
    float sumObjM1;           // sum (conf-1)^2 over obj cells
    float locSum;             // sum smooth-L1 over obj cells
    float nmeSum;             // sum dist/sqrt(wh) over obj cells
};

__device__ __forceinline__ float iou_f(float cx1, float cy1, float w1, float h1,
                                       float cx2, float cy2, float w2, float h2) {
    float x1a = cx1 - w1 * 0.5f, x2a = cx1 + w1 * 0.5f;
    float y1a = cy1 - h1 * 0.5f, y2a = cy1 + h1 * 0.5f;
    float x1b = cx2 - w2 * 0.5f, x2b = cx2 + w2 * 0.5f;
    float y1b = cy2 - h2 * 0.5f, y2b = cy2 + h2 * 0.5f;
    float iw = fminf(x2a, x2b) - fmaxf(x1a, x1b) + 1.0f;
    float ih = fminf(y2a, y2b) - fmaxf(y1a, y1b) + 1.0f;
    float inter = iw * ih;
    float a1 = (x2a - x1a + 1.0f) * (y2a - y1a + 1.0f);
    float a2 = (x2b - x1b + 1.0f) * (y2b - y1b + 1.0f);
    return inter / (a1 + a2 - inter + 1e-16f);
}

__global__ void k_init(Ws* ws) {
    ws->objCount = 0;
    ws->sumAll = 0.0f; ws->sumObjSq = 0.0f; ws->sumObjM1 = 0.0f;
    ws->locSum = 0.0f; ws->nmeSum = 0.0f;
}

// One block of 256 threads: per-target preparation + obj-cell accumulation.
__global__ void k_targets(const float* __restrict__ bboxPred,
                          const float* __restrict__ bboxTgt,
                          Ws* ws) {
    const float AW[AN] = {0.24f, 0.12f, 0.08f, 0.28f, 0.15f};
    __shared__ int slin[BB * TT];
    int s = threadIdx.x;                 // s = b*T + t
    int b = s >> 3, t = s & 7;
    const float* tg = bboxTgt + s * 4;
    float x = tg[0], y = tg[1], w = tg[2], h = tg[3];
    bool valid = (x + y + w + h) != 0.0f;
    float gx = x * (float)GG, gy = y * (float)GG;
    int gi = (int)floorf(gx), gj = (int)floorf(gy);
    gi = min(max(gi, 0), GG - 1); gj = min(max(gj, 0), GG - 1);
    float fx = gx - floorf(gx), fy = gy - floorf(gy);

    // anchor argmax (first max wins, like jnp.argmax)
    float best = -1e30f; int bestA = 0;
    #pragma unroll
    for (int a = 0; a < AN; ++a) {
        float v = iou_f(fx, fy, w, h, 0.5f, 0.5f, AW[a], AW[a]);
        if (v > best) { best = v; bestA = a; }
    }
    int lin = ((b * GG + gj) * GG + gi) * AN + bestA;
    if (!valid) lin = NN;
    slin[s] = lin;
    ws->lin[s] = lin;
    float lb0 = log1pf(fx), lb1 = log1pf(fy), lb2 = log1pf(w), lb3 = log1pf(h);
    ws->logbox[s * 4 + 0] = lb0; ws->logbox[s * 4 + 1] = lb1;
    ws->logbox[s * 4 + 2] = lb2; ws->logbox[s * 4 + 3] = lb3;
    __syncthreads();

    // winner: valid and no later target in same batch maps to same cell
    bool win = valid;
    if (valid) {
        for (int t2 = t + 1; t2 < TT; ++t2)
            if (slin[(b << 3) + t2] == lin) { win = false; break; }
    }
    ws->winner[s] = win ? 1 : 0;

    if (win) {
        atomicAdd(&ws->objCount, 1);
        const float* pp = bboxPred + (size_t)lin * 5;
        float conf = pp[4];
        atomicAdd(&ws->sumObjSq, conf * conf);
        atomicAdd(&ws->sumObjM1, (conf - 1.0f) * (conf - 1.0f));
        float lb[4] = {lb0, lb1, lb2, lb3};
        float sl = 0.0f;
        #pragma unroll
        for (int j = 0; j < 4; ++j) {
            float d = pp[j] - lb[j];
            float ad = fabsf(d);
            sl += (ad < 1.0f) ? 0.5f * d * d : ad - 0.5f;
        }
        atomicAdd(&ws->locSum, sl);
    }
}

// NME via WMMA: each wave handles 16 target slots; dist^2 accumulated as
// diag(A x A^T) with V_WMMA_F32_16X16X4_F32 over 34 K=4 chunks (fully
// unrolled so the scheduler can clause the global_load_b64s ahead of the
// serial WMMA accumulate chain).
// A (16x4 f32) per-lane layout == B (4x16 f32) layout when B = A^T, so the
// same v2f register pair feeds both operands.
__global__ void k_nme(const float* __restrict__ lmPred,
                      const float* __restrict__ lmTgt,
                      Ws* ws) {
    int lane = threadIdx.x & 31;
    int wave = threadIdx.x >> 5;
    int base = blockIdx.x * 128 + wave * 16;     // 2 blocks x 8 waves x 16 slots
    int mslot = base + (lane & 15);
    int koff = (lane >> 4) << 1;                 // lanes 0-15: K 0,1 ; 16-31: K 2,3

    int wflag = ws->winner[mslot];
    int cell = wflag ? ws->lin[mslot] : 0;
    float m = wflag ? 1.0f : 0.0f;
    const float* pp = lmPred + (size_t)cell * LM + koff;
    const float* pt = lmTgt + (size_t)mslot * LM + koff;

    v8f c = {0.0f, 0.0f, 0.0f, 0.0f, 0.0f, 0.0f, 0.0f, 0.0f};
    #pragma unroll
    for (int k0 = 0; k0 < LM; k0 += 4) {
        v2f a;
        a.x = (pp[k0]     - pt[k0])     * m;
        a.y = (pp[k0 + 1] - pt[k0 + 1]) * m;
        // D = A * A^T + C  (diag holds per-slot partial dot products)
        c = __builtin_amdgcn_wmma_f32_16x16x4_f32(false, a, false, a,
                                                  (short)0, c, false, false);
    }

    float carr[8];
    #pragma unroll
    for (int i = 0; i < 8; ++i) carr[i] = c[i];

    // diagonal: D[i][i] at (vgpr i, lane i) for i<8; (vgpr i-8, lane i+16) else
    int slotIdx = -1; float dist2 = 0.0f;
    if (lane < 8)       { slotIdx = base + lane;        dist2 = carr[lane]; }
    else if (lane >= 24){ slotIdx = base + (lane - 16); dist2 = carr[lane - 24]; }

    if (slotIdx >= 0 && ws->winner[slotIdx]) {
        float wh = ws->logbox[slotIdx * 4 + 2] * ws->logbox[slotIdx * 4 + 3];
        atomicAdd(&ws->nmeSum, sqrtf(dist2) * rsqrtf(wh));
    }
}

// Grid-stride reduction of conf^2 over all N cells, vectorized: 4 cells span
// 20 floats = 5 float4s; confs sit at [1].x, [2].y, [3].z, [4].w -> four
// coalesced global_load_b128 per 4 cells.
__global__ void k_confsum(const float* __restrict__ bboxPred, Ws* ws) {
    __shared__ float red[256];
    int tid = threadIdx.x;
    const float4* p4 = (const float4*)bboxPred;
    const int nChunks = NN / 4;                  // 51840 exactly
    float p = 0.0f;
    for (size_t i = (size_t)blockIdx.x * blockDim.x + tid; i < (size_t)nChunks;
         i += (size_t)gridDim.x * blockDim.x) {
        const float4* basep = p4 + i * 5;
        float4 a = basep[1];
        float4 b = basep[2];
        float4 cc = basep[3];
        float4 d = basep[4];
        p += a.x * a.x + b.y * b.y + cc.z * cc.z + d.w * d.w;
    }
    red[tid] = p;
    __syncthreads();
    for (int off = 128; off > 0; off >>= 1) {
        if (tid < off) red[tid] += red[tid + off];
        __syncthreads();
    }
    if (tid == 0) atomicAdd(&ws->sumAll, red[0]);
}

__global__ void k_final(const Ws* ws, float* out) {
    int cnt = ws->objCount;
    float nobj = fmaxf((float)cnt, 1.0f);
    float n_noobj = fmaxf((float)(NN - cnt), 1.0f);
    out[0] = ws->nmeSum / (IMG * 68.0f * nobj);
    out[1] = L_COOR * ws->locSum / (nobj * 4.0f);
    out[2] = L_NOOBJ * (ws->sumAll - ws->sumObjSq) / n_noobj + ws->sumObjM1 / nobj;
}

// best_pred: per (b,t), IoU of log_box vs the 5 anchor predictions at its cell.
__global__ void k_best(const float* __restrict__ bboxPred,
                       const float* __restrict__ bboxTgt,
                       float* out) {
    int s = threadIdx.x;
    int b = s >> 3;
    const float* tg = bboxTgt + s * 4;
    float x = tg[0], y = tg[1], w = tg[2], h = tg[3];
    bool valid = (x + y + w + h) != 0.0f;
    float gx = x * (float)GG, gy = y * (float)GG;
    int gi = min(max((int)floorf(gx), 0), GG - 1);
    int gj = min(max((int)floorf(gy), 0), GG - 1);
    float fx = gx - floorf(gx), fy = gy - floorf(gy);
    float lb0 = log1pf(fx), lb1 = log1pf(fy), lb2 = log1pf(w), lb3 = log1pf(h);

    size_t cb = (size_t)(((b * GG + gj) * GG + gi) * AN) * 5;
    float best = -1e30f; int bestA = 0;
    #pragma unroll
    for (int a = 0; a < AN; ++a) {
        const float* pr = bboxPred + cb + (size_t)a * 5;
        float v = iou_f(lb0, lb1, lb2, lb3, pr[0], pr[1], pr[2], pr[3]);
        if (v > best) { best = v; bestA = a; }
    }
    const float* pr = bboxPred + cb + (size_t)bestA * 5;
    float cx = pr[0], cy = pr[1], bw = pr[2], bh = pr[3], cf = pr[4];
    float vmask = valid ? 1.0f : 0.0f;
    float* o = out + 3 + (size_t)s * 5;
    o[0] = (cx - bw * 0.5f) * vmask;
    o[1] = (cx + bw * 0.5f) * vmask;
    o[2] = (cy - bh * 0.5f) * vmask;
    o[3] = (cy + bh * 0.5f) * vmask;
    o[4] = cf * vmask;
}

extern "C" void kernel_launch(void* const* d_in, const int* in_sizes, int n_in,
                              void* d_out, int out_size, void* d_ws, size_t ws_size,
                              hipStream_t stream) {
    const float* bboxPred = (const float*)d_in[0];   // (32,36,36,5,5)
    const float* lmPred   = (const float*)d_in[1];   // (32,36,36,5,68,2)
    const float* bboxTgt  = (const float*)d_in[2];   // (32,8,4)
    const float* lmTgt    = (const float*)d_in[3];   // (32,8,68,2)
    float* out = (float*)d_out;                      // [nme, loc, conf, best_pred(32,8,5)]
    Ws* ws = (Ws*)d_ws;

    k_init<<<1, 1, 0, stream>>>(ws);
    k_targets<<<1, 256, 0, stream>>>(bboxPred, bboxTgt, ws);
    k_nme<<<2, 256, 0, stream>>>(lmPred, lmTgt, ws);
    k_confsum<<<128, 256, 0, stream>>>(bboxPred, ws);
    k_final<<<1, 1, 0, stream>>>(ws, out);
    k_best<<<1, 256, 0, stream>>>(bboxPred, bboxTgt, out);
}